// SightAndNearLoss_10015863734569
// MI455X (gfx1250) — compile-verified
//
#include <hip/hip_runtime.h>
#include <hip/hip_bf16.h>
#include <stdint.h>

#define EPSILON 0.05f
#define WAVES_PER_BLOCK 8
#define THREADS 256
#define CHUNK 128   // floats per async chunk per array (32 lanes x float4)
#define NBUF 2

typedef __attribute__((ext_vector_type(2))) float v2f;
typedef __attribute__((ext_vector_type(8))) float v8f;

// ---- CDNA5 async global->LDS staging (ASYNCcnt path) ----------------------
__device__ __forceinline__ void async_load_b128(unsigned lds_byte_addr, const void* gptr) {
    asm volatile("global_load_async_to_lds_b128 %0, %1, off"
                 :
                 : "v"(lds_byte_addr), "v"((unsigned long long)(uintptr_t)gptr)
                 : "memory");
}
__device__ __forceinline__ void wait_async_le2() {
    asm volatile("s_wait_asynccnt 0x2" ::: "memory");
}
__device__ __forceinline__ void wait_async_0() {
    asm volatile("s_wait_asynccnt 0x0" ::: "memory");
}
__device__ __forceinline__ void wait_ds_0() {
    asm volatile("s_wait_dscnt 0x0" ::: "memory");
}

// ---------------------------------------------------------------------------
// Kernel 1: one wave32 per ray. Streams z/w rows through LDS double buffers
// using async-to-LDS B128 gathers (512B per wave per instruction), classifies
// each sample against [d-eps, d+eps), wave-reduces, accumulates per-block
// partial triples (sum_empty, sum_near, mask_count) deterministically.
// ---------------------------------------------------------------------------
__global__ __launch_bounds__(THREADS)
void sight_near_partials(const float* __restrict__ ray_depth,
                         const float* __restrict__ z_vals,
                         const float* __restrict__ weights,
                         const unsigned char* __restrict__ ray_mask,
                         float* __restrict__ ws,
                         int n_rays, int n_samples)
{
    // [wave][buf][arr(z=0,w=1)][CHUNK]
    __shared__ __align__(16) float smem[WAVES_PER_BLOCK * NBUF * 2 * CHUNK];
    __shared__ float wave_part[WAVES_PER_BLOCK][3];

    const int lane = threadIdx.x & 31;
    const int wave = threadIdx.x >> 5;
    const int nchunks = n_samples / CHUNK;

    float acc_e = 0.f, acc_near = 0.f, acc_cnt = 0.f;

    for (int r = blockIdx.x * WAVES_PER_BLOCK + wave; r < n_rays;
         r += gridDim.x * WAVES_PER_BLOCK) {

        const float d  = ray_depth[r];
        const float lo = d - EPSILON;
        const float hi = d + EPSILON;
        const float* zrow = z_vals  + (size_t)r * n_samples;
        const float* wrow = weights + (size_t)r * n_samples;

        auto issue = [&](int c) {
            const int buf = c & (NBUF - 1);
            const float* lz = &smem[((wave * NBUF + buf) * 2 + 0) * CHUNK + lane * 4];
            const float* lw = &smem[((wave * NBUF + buf) * 2 + 1) * CHUNK + lane * 4];
            // ensure any in-flight LDS reads of this buffer completed
            wait_ds_0();
            async_load_b128((unsigned)(uintptr_t)lz, zrow + c * CHUNK + lane * 4);
            async_load_b128((unsigned)(uintptr_t)lw, wrow + c * CHUNK + lane * 4);
        };

        issue(0);
        float e = 0.f, nsum = 0.f;

        for (int c = 0; c < nchunks; ++c) {
            if (c + 1 < nchunks) {
                issue(c + 1);
                wait_async_le2();   // oldest chunk (c) has landed in LDS
            } else {
                wait_async_0();
            }
            const int buf = c & (NBUF - 1);
            const float4 z4 = *reinterpret_cast<const float4*>(
                &smem[((wave * NBUF + buf) * 2 + 0) * CHUNK + lane * 4]);
            const float4 w4 = *reinterpret_cast<const float4*>(
                &smem[((wave * NBUF + buf) * 2 + 1) * CHUNK + lane * 4]);

            const float zz[4] = {z4.x, z4.y, z4.z, z4.w};
            const float wv[4] = {w4.x, w4.y, w4.z, w4.w};
#pragma unroll
            for (int k = 0; k < 4; ++k) {
                const bool below  = zz[k] < lo;
                const bool inband = (zz[k] < hi) && !below;
                e    += below  ? wv[k] * wv[k] : 0.f;
                nsum += inband ? wv[k]         : 0.f;
            }
        }

        // wave32 reduction (butterfly)
#pragma unroll
        for (int off = 16; off > 0; off >>= 1) {
            e    += __shfl_xor(e,    off, 32);
            nsum += __shfl_xor(nsum, off, 32);
        }

        const float m  = ray_mask[r] ? 1.f : 0.f;
        const float dn = 1.f - nsum;
        acc_e    += m * e;
        acc_near += m * dn * dn;
        acc_cnt  += m;
    }

    if (lane == 0) {
        wave_part[wave][0] = acc_e;
        wave_part[wave][1] = acc_near;
        wave_part[wave][2] = acc_cnt;
    }
    __syncthreads();
    if (threadIdx.x == 0) {
        float E = 0.f, N = 0.f, C = 0.f;
#pragma unroll
        for (int w = 0; w < WAVES_PER_BLOCK; ++w) {
            E += wave_part[w][0];
            N += wave_part[w][1];
            C += wave_part[w][2];
        }
        ws[(size_t)blockIdx.x * 3 + 0] = E;
        ws[(size_t)blockIdx.x * 3 + 1] = N;
        ws[(size_t)blockIdx.x * 3 + 2] = C;
    }
}

// ---------------------------------------------------------------------------
// WMMA-based wave32 sum: A(16x4) holds {x,0} per lane, B = all ones, so
// D[m][n] = A[m][0..3] row-sum = x_m + x_{m+16}. Summing the 8 accumulator
// VGPRs gives each lane half-wave partials; one shfl_xor(16) completes the
// full 32-lane sum. EXEC is all ones here (no divergence before the call).
// ---------------------------------------------------------------------------
__device__ __forceinline__ float wmma_wave_sum(float x) {
    v2f a; a.x = x;   a.y = 0.f;
    v2f b; b.x = 1.f; b.y = 1.f;
    v8f c = {};
    c = __builtin_amdgcn_wmma_f32_16x16x4_f32(false, a, false, b,
                                              (short)0, c, false, false);
    float s = c[0] + c[1] + c[2] + c[3] + c[4] + c[5] + c[6] + c[7];
    s += __shfl_xor(s, 16, 32);
    return s;
}

__global__ __launch_bounds__(THREADS)
void sight_near_finalize(const float* __restrict__ ws,
                         float* __restrict__ out, int nblocks)
{
    __shared__ float red[WAVES_PER_BLOCK][3];
    const int t = threadIdx.x;

    float e = 0.f, nr = 0.f, ct = 0.f;
    for (int i = t; i < nblocks; i += THREADS) {
        e  += ws[(size_t)i * 3 + 0];
        nr += ws[(size_t)i * 3 + 1];
        ct += ws[(size_t)i * 3 + 2];
    }

    const float we = wmma_wave_sum(e);
    const float wn = wmma_wave_sum(nr);
    const float wc = wmma_wave_sum(ct);

    const int wave = t >> 5, lane = t & 31;
    if (lane == 0) { red[wave][0] = we; red[wave][1] = wn; red[wave][2] = wc; }
    __syncthreads();
    if (t == 0) {
        float E = 0.f, N = 0.f, C = 0.f;
#pragma unroll
        for (int w = 0; w < WAVES_PER_BLOCK; ++w) {
            E += red[w][0]; N += red[w][1]; C += red[w][2];
        }
        out[0] = E / C;   // loss_empty
        out[1] = N / C;   // loss_near
    }
}

// ---------------------------------------------------------------------------
extern "C" void kernel_launch(void* const* d_in, const int* in_sizes, int n_in,
                              void* d_out, int out_size, void* d_ws, size_t ws_size,
                              hipStream_t stream)
{
    const float*         ray_depth = (const float*)d_in[0];          // [N,1] f32
    const float*         z_vals    = (const float*)d_in[1];          // [N,S] f32
    const float*         weights   = (const float*)d_in[2];          // [N,S] f32
    const unsigned char* ray_mask  = (const unsigned char*)d_in[3];  // [N] bool (1B)
    float*               out       = (float*)d_out;                  // 2 f32 scalars
    float*               ws        = (float*)d_ws;

    const int n_rays    = in_sizes[0];              // 16384
    const int n_samples = in_sizes[1] / n_rays;     // 1024

    const int blocks = (n_rays + WAVES_PER_BLOCK - 1) / WAVES_PER_BLOCK; // 2048

    sight_near_partials<<<blocks, THREADS, 0, stream>>>(
        ray_depth, z_vals, weights, ray_mask, ws, n_rays, n_samples);
    sight_near_finalize<<<1, THREADS, 0, stream>>>(ws, out, blocks);
}